// EdgeModel_out_11227044512393
// MI455X (gfx1250) — compile-verified
//
#include <hip/hip_runtime.h>

typedef float v2f __attribute__((ext_vector_type(2)));
typedef float v8f __attribute__((ext_vector_type(8)));

#define E_TOTAL     4000000
#define FEAT_STRIDE 38          // 36 used (K padded to 36), +2 pad, keeps 8B row align
#define H_STRIDE    18          // 8 used (K=5 padded to 8), keeps 8B row align
#define WAVE_LDS    (32*FEAT_STRIDE + 32*H_STRIDE + 160)   // 1952 dwords / wave
#define WAVES_PER_BLOCK 8

static __device__ __forceinline__ void lds_fence_wave() {
    // LDS ops from one wave complete in order; this drains DScnt and stops
    // the compiler from hoisting ds_loads above the cross-lane ds_stores.
    asm volatile("s_wait_dscnt 0" ::: "memory");
}

__global__ __launch_bounds__(256) void edge_mlp_wmma_kernel(
    const float* __restrict__ x_s,       // [N_S,10]
    const float* __restrict__ x_t,       // [N_T,5]
    const int*   __restrict__ src,       // [E]
    const int*   __restrict__ tgt,       // [E]
    const float* __restrict__ edge_attr, // [E,10]
    const float* __restrict__ u,         // [B,10]
    const int*   __restrict__ batch_e,   // [E]
    const float* __restrict__ W1,        // [35,5] row-major
    const float* __restrict__ b1,        // [5]
    const float* __restrict__ W2,        // [5,5]
    const float* __restrict__ b2,        // [5]
    float*       __restrict__ out)       // [E,5]
{
    __shared__ float smem[WAVES_PER_BLOCK * WAVE_LDS];

    const int lane  = threadIdx.x & 31;
    const int wave  = threadIdx.x >> 5;
    const int n     = lane & 15;   // N (or M) index within 16
    const int khalf = lane >> 4;   // which K-half this lane holds

    float* feat = &smem[wave * WAVE_LDS];
    float* hbuf = feat + 32 * FEAT_STRIDE;
    float* obuf = hbuf + 32 * H_STRIDE;

    // ---- preload W1 into 9 B-fragments (K = 0..35, zero-padded), W2 into 2 ----
    // B 4x16 f32 layout (mirror of documented A layout):
    //   element (k, n) -> lane = n + 16*(k>>1), vgpr = k&1
    v2f w1b[9];
#pragma unroll
    for (int kb = 0; kb < 9; ++kb) {
#pragma unroll
        for (int r = 0; r < 2; ++r) {
            const int K = kb * 4 + khalf * 2 + r;
            float v = 0.0f;
            if (n < 5 && K < 35) v = W1[K * 5 + n];
            w1b[kb][r] = v;
        }
    }
    v2f w2b[2];
#pragma unroll
    for (int kb = 0; kb < 2; ++kb) {
#pragma unroll
        for (int r = 0; r < 2; ++r) {
            const int K = kb * 4 + khalf * 2 + r;
            float v = 0.0f;
            if (n < 5 && K < 5) v = W2[K * 5 + n];
            w2b[kb][r] = v;
        }
    }
    const float bias1 = (n < 5) ? b1[n] : 0.0f;  // zero padding cols => h cols 5..15 stay 0
    const float bias2 = (n < 5) ? b2[n] : 0.0f;

    const int waveId = blockIdx.x * WAVES_PER_BLOCK + wave;
    const int nWaves = gridDim.x * WAVES_PER_BLOCK;

    // E_TOTAL % 32 == 0 -> every wave iteration is a full 32-edge group,
    // EXEC is all-ones at every WMMA.
    for (int e0 = waveId * 32; e0 < E_TOTAL; e0 += nWaves * 32) {
        // ================= gather: lane L owns edge e0+L =================
        const int e = e0 + lane;
        const int s = __builtin_nontemporal_load(src + e);
        const int t = __builtin_nontemporal_load(tgt + e);
        const int b = __builtin_nontemporal_load(batch_e + e);

        float* row = feat + lane * FEAT_STRIDE;

        const v2f* ps = (const v2f*)(x_s + s * 10);      // 8B aligned (s*40)
#pragma unroll
        for (int i = 0; i < 5; ++i) ((v2f*)row)[i] = ps[i];         // feat[0..9]
#pragma unroll
        for (int i = 0; i < 5; ++i) row[10 + i] = x_t[t * 5 + i];   // feat[10..14]
        {
            const v2f* pe = (const v2f*)(edge_attr + e * 10);       // 8B aligned
#pragma unroll
            for (int i = 0; i < 5; ++i) {
                v2f ea = __builtin_nontemporal_load(pe + i);
                row[15 + 2 * i]     = ea.x;                          // feat[15..24]
                row[15 + 2 * i + 1] = ea.y;
            }
        }
        {
            const v2f* pu = (const v2f*)(u + b * 10);               // 8B aligned
#pragma unroll
            for (int i = 0; i < 5; ++i) {
                v2f uv = pu[i];
                row[25 + 2 * i]     = uv.x;                          // feat[25..34]
                row[25 + 2 * i + 1] = uv.y;
            }
        }
        row[35] = 0.0f;   // K padding for last K-block

        lds_fence_wave();

        // ================= layer 1 + leaky relu (two 16-edge tiles) ======
#pragma unroll
        for (int T = 0; T < 2; ++T) {
            v8f acc;
#pragma unroll
            for (int i = 0; i < 8; ++i) acc[i] = bias1;

            // A 16x4 f32 layout: element (m,k) -> lane = m + 16*(k>>1), vgpr = k&1
            const float* arow = feat + (T * 16 + n) * FEAT_STRIDE + khalf * 2;
#pragma unroll
            for (int kb = 0; kb < 9; ++kb) {
                v2f a = *(const v2f*)(arow + kb * 4);                // ds_load_b64
                acc = __builtin_amdgcn_wmma_f32_16x16x4_f32(
                          false, a, false, w1b[kb], (short)0, acc, false, false);
            }
            // leaky_relu(x, 0.1) == max(x, 0.1*x)
#pragma unroll
            for (int i = 0; i < 8; ++i) acc[i] = fmaxf(acc[i], 0.1f * acc[i]);

            // D layout: vgpr r holds (m = T*16 + r + 8*khalf, col = n).
            // Stage h row-major so layer-2 can read it in A layout; cols 5..7 are 0.
            if (n < 8) {
#pragma unroll
                for (int r = 0; r < 8; ++r) {
                    const int m = T * 16 + r + 8 * khalf;
                    hbuf[m * H_STRIDE + n] = acc[r];
                }
            }
        }

        lds_fence_wave();

        // ================= layer 2 =======================================
#pragma unroll
        for (int T = 0; T < 2; ++T) {
            v8f acc;
#pragma unroll
            for (int i = 0; i < 8; ++i) acc[i] = bias2;

            const float* arow = hbuf + (T * 16 + n) * H_STRIDE + khalf * 2;
#pragma unroll
            for (int kb = 0; kb < 2; ++kb) {
                v2f a = *(const v2f*)(arow + kb * 4);
                acc = __builtin_amdgcn_wmma_f32_16x16x4_f32(
                          false, a, false, w2b[kb], (short)0, acc, false, false);
            }
            // Stage outputs linearly (m*5 + n) for coalesced global writes.
            if (n < 5) {
#pragma unroll
                for (int r = 0; r < 8; ++r) {
                    const int m = T * 16 + r + 8 * khalf;
                    obuf[m * 5 + n] = acc[r];
                }
            }
        }

        lds_fence_wave();

        // ============ coalesced output: 160 contiguous dwords ============
#pragma unroll
        for (int j = 0; j < 5; ++j) {
            __builtin_nontemporal_store(obuf[j * 32 + lane],
                                        out + e0 * 5 + j * 32 + lane);
        }
    }
}

extern "C" void kernel_launch(void* const* d_in, const int* in_sizes, int n_in,
                              void* d_out, int out_size, void* d_ws, size_t ws_size,
                              hipStream_t stream) {
    const float* x_s       = (const float*)d_in[0];
    const float* x_t       = (const float*)d_in[1];
    const int*   edge_idx  = (const int*)d_in[2];   // [2, E]
    const float* edge_attr = (const float*)d_in[3];
    const float* u         = (const float*)d_in[4];
    const int*   batch_e   = (const int*)d_in[5];
    const float* W1        = (const float*)d_in[6];
    const float* b1        = (const float*)d_in[7];
    const float* W2        = (const float*)d_in[8];
    const float* b2        = (const float*)d_in[9];
    float*       out       = (float*)d_out;

    const int E = in_sizes[5];                // 4,000,000
    const int* src = edge_idx;
    const int* tgt = edge_idx + E;

    dim3 block(256);
    dim3 grid(2048);                          // persistent-ish grid-stride
    edge_mlp_wmma_kernel<<<grid, block, 0, stream>>>(
        x_s, x_t, src, tgt, edge_attr, u, batch_e, W1, b1, W2, b2, out);
}